// RuleLearner_85925115724202
// MI455X (gfx1250) — compile-verified
//
#include <hip/hip_runtime.h>
#include <hip/hip_bf16.h>
#include <math.h>

// Problem dims (fixed by reference setup_inputs)
#define BSZ 32
#define NI  4     // invariants I
#define NL  4     // nodes per invariant L
#define NN  96    // graph nodes N
#define NP1 69
#define NP2 8
#define NSYM 64
#define NTHREADS 384
// sm_scale = log(0.999*3/0.001) = log(2997)
#define SM_SCALE 8.0053673f

typedef __attribute__((ext_vector_type(2))) float v2f;
typedef __attribute__((ext_vector_type(8))) float v8f;

// LDS layout (float offsets)
#define OFF_UF   0        // unary_feats[b]: [96][69]            6624
#define OFF_C1   6624     // gated unary conds [I][L][69]        1104
#define OFF_C2   7728     // gated binary conds [I][L][L][8]      512 (32B aligned)
#define OFF_UM   8240     // unary_match [I*L][96]               1536
#define OFF_NS0  9776     // node_select0 [I*L][96]              1536
#define OFF_PAIR 11312    // pair [48][96] (l!=k combos)         4608
#define OFF_UNI  15920    // uni_sets [I*L][96]                  1536
#define OFF_NS   17456    // node_select [I*L][96]               1536
#define OFF_D    18992    // GEMM result D [16][64]              1024
#define OFF_OM   20016    // om softmax [I][5]                     20
#define OFF_PS   20036    // probsum per (i,l)                     16
#define OFF_SEL  20052    // inv_select[i]                          4
#define LDS_FLOATS 20056

__device__ __forceinline__ float sigf(float x) { return 1.0f / (1.0f + expf(-x)); }

__device__ void softmax_row(const float* in, float* out, int len, float scale) {
    float mx = -INFINITY;
    for (int j = 0; j < len; ++j) mx = fmaxf(mx, in[j] * scale);
    float s = 0.0f;
    for (int j = 0; j < len; ++j) { float e = expf(in[j] * scale - mx); out[j] = e; s += e; }
    float inv = 1.0f / s;
    for (int j = 0; j < len; ++j) out[j] *= inv;
}

__global__ __launch_bounds__(NTHREADS)
void rule_learner_fused(const float* __restrict__ inv_unary_feats,   // (I,L,P1)
                        const float* __restrict__ inv_binary_feats,  // (I,L,L,P2)
                        const float* __restrict__ unary_feats,       // (B,N,P1)
                        const float* __restrict__ binary_feats,      // (B,N,N,P2)
                        const float* __restrict__ inv_unaryp,        // (I,L,P1)
                        const float* __restrict__ inv_binaryp,       // (I,L,L,P2)
                        const float* __restrict__ inv_out_map,       // (I,L+1)
                        const int*   __restrict__ inv_label,         // (I,)
                        float* __restrict__ out)                     // (B,NSYM)
{
    extern __shared__ float smem[];
    const int b   = blockIdx.x;
    const int tid = threadIdx.x;

    // ---------- Stage 0: stage unary_feats[b] to LDS; gated conds; om ----------
    for (int idx = tid; idx < NN * NP1; idx += NTHREADS)
        smem[OFF_UF + idx] = unary_feats[(size_t)b * NN * NP1 + idx];
    for (int idx = tid; idx < NI * NL * NP1; idx += NTHREADS)
        smem[OFF_C1 + idx] = sigf(inv_unaryp[idx]) * inv_unary_feats[idx];
    for (int idx = tid; idx < NI * NL * NL * NP2; idx += NTHREADS)
        smem[OFF_C2 + idx] = sigf(inv_binaryp[idx]) * inv_binary_feats[idx];
    if (tid < NI)
        softmax_row(inv_out_map + tid * (NL + 1), &smem[OFF_OM + tid * (NL + 1)], NL + 1, 1.0f);
    __syncthreads();

    // ---------- Stage 1: unary_match[il][n] = prod_p (1 - c1*(1-uf)) ----------
    for (int idx = tid; idx < NI * NL * NN; idx += NTHREADS) {
        const int il = idx / NN, n = idx % NN;
        const float* c1 = &smem[OFF_C1 + il * NP1];
        const float* uf = &smem[OFF_UF + n * NP1];
        float prod = 1.0f;
        for (int p = 0; p < NP1; ++p) prod *= (1.0f - c1[p] * (1.0f - uf[p]));
        smem[OFF_UM + il * NN + n] = prod;
    }
    __syncthreads();

    // ---------- Stage 2: ns0 = softmax(um * s) over n; zero pair buffer ----------
    if (tid < NI * NL)
        softmax_row(&smem[OFF_UM + tid * NN], &smem[OFF_NS0 + tid * NN], NN, SM_SCALE);
    for (int idx = tid; idx < 48 * NN; idx += NTHREADS) smem[OFF_PAIR + idx] = 0.0f;
    __syncthreads();

    // ---------- Stage 3: pair[i,l,k,n] = sum_m ns0[i,k,m] * prod_p(1 - c2*(1-f2)) ----------
    {
        const int n = tid >> 2;        // 0..95
        const int q = tid & 3;         // m-quarter
        const float* f2base = binary_feats + (((size_t)b * NN + n) * NN) * NP2;
        float acc[48];
#pragma unroll
        for (int c = 0; c < 48; ++c) acc[c] = 0.0f;

        for (int mi = 0; mi < 24; ++mi) {
            const int m = q * 24 + mi;
            const float4* fp = (const float4*)(f2base + m * NP2);
            const float4 fa = fp[0], fb = fp[1];
            const float t0 = 1.0f - fa.x, t1 = 1.0f - fa.y, t2 = 1.0f - fa.z, t3 = 1.0f - fa.w;
            const float t4 = 1.0f - fb.x, t5 = 1.0f - fb.y, t6 = 1.0f - fb.z, t7 = 1.0f - fb.w;
#pragma unroll
            for (int i = 0; i < 4; ++i) {
                float nsk[4];
#pragma unroll
                for (int k = 0; k < 4; ++k) nsk[k] = smem[OFF_NS0 + (i * 4 + k) * NN + m];
#pragma unroll
                for (int l = 0; l < 4; ++l) {
#pragma unroll
                    for (int k = 0; k < 4; ++k) {
                        if (k == l) continue;
                        const int j = l * 3 + (k > l ? k - 1 : k);
                        const float4* cp = (const float4*)&smem[OFF_C2 + (i * 16 + l * 4 + k) * NP2];
                        const float4 ca = cp[0], cb = cp[1];
                        float p01 = (1.0f - ca.x * t0) * (1.0f - ca.y * t1);
                        float p23 = (1.0f - ca.z * t2) * (1.0f - ca.w * t3);
                        float p45 = (1.0f - cb.x * t4) * (1.0f - cb.y * t5);
                        float p67 = (1.0f - cb.z * t6) * (1.0f - cb.w * t7);
                        acc[i * 12 + j] += nsk[k] * ((p01 * p23) * (p45 * p67));
                    }
                }
            }
        }
#pragma unroll
        for (int c = 0; c < 48; ++c) atomicAdd(&smem[OFF_PAIR + c * NN + n], acc[c]);
    }
    __syncthreads();

    // ---------- Stage 4: uni[il][n] = um * prod_{k!=l} pair ----------
    for (int idx = tid; idx < NI * NL * NN; idx += NTHREADS) {
        const int il = idx / NN, n = idx % NN;
        const int i = il >> 2, l = il & 3;
        float v = smem[OFF_UM + il * NN + n];
#pragma unroll
        for (int k = 0; k < 4; ++k) {
            if (k == l) continue;
            const int j = l * 3 + (k > l ? k - 1 : k);
            v *= smem[OFF_PAIR + (i * 12 + j) * NN + n];
        }
        smem[OFF_UNI + il * NN + n] = v;
    }
    __syncthreads();

    // ---------- Stage 5: ns = softmax(uni*s); probsum per (i,l) ----------
    if (tid < NI * NL) {
        softmax_row(&smem[OFF_UNI + tid * NN], &smem[OFF_NS + tid * NN], NN, SM_SCALE);
        float qq = 1.0f;
        for (int n = 0; n < NN; ++n) qq *= (1.0f - smem[OFF_UNI + tid * NN + n]);
        smem[OFF_PS + tid] = 1.0f - qq;
    }
    __syncthreads();

    // ---------- Stage 6: wave0 does WMMA GEMM D[16][64] = (om*ns)[16][96] x uf[96][64];
    //                     one other lane computes inv_select ----------
    if (tid == 32) {
        float iu[NI], mx = -INFINITY;
#pragma unroll
        for (int i = 0; i < NI; ++i) {
            float v = 1.0f;
#pragma unroll
            for (int l = 0; l < NL; ++l) v *= smem[OFF_PS + i * 4 + l];
            iu[i] = v * SM_SCALE;
            mx = fmaxf(mx, iu[i]);
        }
        float s = 0.0f, e[NI];
#pragma unroll
        for (int i = 0; i < NI; ++i) { e[i] = expf(iu[i] - mx); s += e[i]; }
#pragma unroll
        for (int i = 0; i < NI; ++i) smem[OFF_SEL + i] = e[i] / s;
    }
    if (tid < 32) {
        const int lane = tid;
        const int row  = lane & 15;       // M row = i*4 + l
        const int khi  = lane >> 4;       // which K half-pair this lane holds
        const float omr = smem[OFF_OM + (row >> 2) * (NL + 1) + (row & 3)];
#pragma unroll
        for (int ut = 0; ut < 4; ++ut) {
            v8f c = {0.f, 0.f, 0.f, 0.f, 0.f, 0.f, 0.f, 0.f};
            for (int k0 = 0; k0 < NN; k0 += 4) {
                const int kA = k0 + 2 * khi;
                // A: 16x4 f32, lanes 0-15 -> K=k0,k0+1 ; lanes 16-31 -> K=k0+2,k0+3
                v2f a = { omr * smem[OFF_NS + row * NN + kA],
                          omr * smem[OFF_NS + row * NN + kA + 1] };
                // B: 4x16 f32, same per-lane K ordering, N = lane&15
                v2f bv = { smem[OFF_UF + kA * NP1 + 5 + ut * 16 + row],
                           smem[OFF_UF + (kA + 1) * NP1 + 5 + ut * 16 + row] };
                c = __builtin_amdgcn_wmma_f32_16x16x4_f32(
                        false, a, false, bv, (short)0, c, false, false);
            }
            // C/D layout: vgpr j -> M = j + 8*khi, N = lane&15
#pragma unroll
            for (int j = 0; j < 8; ++j)
                smem[OFF_D + (j + 8 * khi) * NSYM + ut * 16 + row] = c[j];
        }
    }
    __syncthreads();

    // ---------- Stage 7: predictions[b][s] ----------
    if (tid < NSYM) {
        const int s = tid;
        float r = 0.0f;
#pragma unroll
        for (int i = 0; i < NI; ++i) {
            float nodes = smem[OFF_D + (i * 4 + 0) * NSYM + s]
                        + smem[OFF_D + (i * 4 + 1) * NSYM + s]
                        + smem[OFF_D + (i * 4 + 2) * NSYM + s]
                        + smem[OFF_D + (i * 4 + 3) * NSYM + s];
            float cst = (inv_label[i] == s) ? smem[OFF_OM + i * (NL + 1) + NL] : 0.0f;
            r += smem[OFF_SEL + i] * (nodes + cst);
        }
        out[b * NSYM + s] = r;
    }
}

extern "C" void kernel_launch(void* const* d_in, const int* in_sizes, int n_in,
                              void* d_out, int out_size, void* d_ws, size_t ws_size,
                              hipStream_t stream) {
    (void)in_sizes; (void)n_in; (void)out_size; (void)d_ws; (void)ws_size;
    const float* inv_unary_feats  = (const float*)d_in[0];
    const float* inv_binary_feats = (const float*)d_in[1];
    const float* unary_feats      = (const float*)d_in[2];
    const float* binary_feats     = (const float*)d_in[3];
    const float* inv_unaryp       = (const float*)d_in[4];
    const float* inv_binaryp      = (const float*)d_in[5];
    const float* inv_out_map      = (const float*)d_in[6];
    const int*   inv_label        = (const int*)d_in[7];
    float* out = (float*)d_out;

    const size_t shmem = (size_t)LDS_FLOATS * sizeof(float);
    rule_learner_fused<<<dim3(BSZ), dim3(NTHREADS), shmem, stream>>>(
        inv_unary_feats, inv_binary_feats, unary_feats, binary_feats,
        inv_unaryp, inv_binaryp, inv_out_map, inv_label, out);
}